// MusicTheoryEncoder_66194035966529
// MI455X (gfx1250) — compile-verified
//
#include <hip/hip_runtime.h>
#include <math.h>

typedef __attribute__((ext_vector_type(2))) float v2f;
typedef __attribute__((ext_vector_type(8))) float v8f;

#define S_LEN 1024

// ---------------------------------------------------------------------------
// Kernel 1: per-row feature extraction. grid = 32 blocks (one per batch row),
// block = 256 threads. Histograms parallel w/ LDS atomics, O(S^2) time-group
// max parallel, serial scans on single lanes of distinct waves.
// ---------------------------------------------------------------------------
__global__ __launch_bounds__(256) void feat_kernel(
    const int* __restrict__ tokens,
    float* __restrict__ harmF, float* __restrict__ melF,
    float* __restrict__ rhyF,  float* __restrict__ dynF) {
  __shared__ int   tok[S_LEN];
  __shared__ int   tms[S_LEN];
  __shared__ int   gmax[S_LEN];
  __shared__ unsigned char focc[S_LEN];
  __shared__ float pc_hist[12];
  __shared__ float vhist[32];
  __shared__ float s_npitch, s_nvel, s_vsum, s_vsumsq;
  __shared__ int   s_vmax, s_vmin;

  const int b   = blockIdx.x;
  const int tid = threadIdx.x;

  // ---- load + init ----
  for (int i = tid; i < S_LEN; i += 256) tok[i] = tokens[b * S_LEN + i];
  if (tid < 12) pc_hist[tid] = 0.f;
  if (tid < 32) vhist[tid]   = 0.f;
  if (tid == 0) { s_npitch = 0.f; s_nvel = 0.f; s_vsum = 0.f; s_vsumsq = 0.f;
                  s_vmax = -1; s_vmin = 1000000000; }
  __syncthreads();

  // ---- parallel histograms (harmony pitch classes, velocity stats) ----
  for (int i = tid; i < S_LEN; i += 256) {
    int t = tok[i];
    if (t >= 506 && t < 594) {
      int pitch = t - 485;                    // t-506+21
      atomicAdd(&pc_hist[pitch % 12], 1.0f);
      atomicAdd(&s_npitch, 1.0f);
    }
    if (t >= 682 && t < 714) {
      int vb = t - 682;
      float v = (float)(vb * 4 + 2);
      atomicAdd(&vhist[vb], 1.0f);
      atomicAdd(&s_nvel, 1.0f);
      atomicAdd(&s_vsum, v);
      atomicAdd(&s_vsumsq, v * v);
      atomicMax(&s_vmax, vb * 4 + 2);
      atomicMin(&s_vmin, vb * 4 + 2);
    }
  }
  __syncthreads();

  // ---- serial time decoding (thread 0) ----
  if (tid == 0) {
    int cur = 0, chunk = 0;
    for (int i = 0; i < S_LEN; ++i) {
      int t = tok[i];
      if (t == 4) chunk = cur;
      if (t >= 5 && t <= 505) cur = chunk + (t - 5) * 10;
      tms[i] = cur;
    }
  }
  __syncthreads();

  // ---- parallel time-group max + first-occurrence (O(S^2) per row) ----
  for (int i = tid; i < S_LEN; i += 256) {
    int ti = tok[i];
    bool p = (ti >= 506 && ti < 594);
    int g = -1;
    bool first = p;
    if (p) {
      int myT = tms[i];
      for (int j = 0; j < S_LEN; ++j) {
        int tj = tok[j];
        if (tj >= 506 && tj < 594 && tms[j] == myT) {
          int pj = tj - 485;
          if (pj > g) g = pj;
          if (j < i) first = false;
        }
      }
    }
    gmax[i] = g;
    focc[i] = first ? 1 : 0;
  }
  __syncthreads();

  // ---- melody serial scan (lane 0, wave 0) ----
  if (tid == 0) {
    float ih[25]; for (int i = 0; i < 25; ++i) ih[i] = 0.f;
    float up = 0.f, down = 0.f, fl = 0.f, n_sky = 0.f, sm = 0.f;
    int mx = -1000000000, mn = 1000000000;
    int prev = 0; bool has = false;
    for (int i = 0; i < S_LEN; ++i) {
      if (focc[i]) {
        int gv = gmax[i];
        if (has) {
          int diff = gv - prev;
          int idx = diff + 12; idx = idx < 0 ? 0 : (idx > 24 ? 24 : idx);
          ih[idx] += 1.f;
          if (diff > 0) up += 1.f; else if (diff < 0) down += 1.f; else fl += 1.f;
        }
        prev = gv; has = true;
        n_sky += 1.f;
        if (gv > mx) mx = gv;
        if (gv < mn) mn = gv;
        sm += (float)gv;
      }
    }
    float tot = 0.f; for (int i = 0; i < 25; ++i) tot += ih[i];
    float dn  = tot > 0.f ? tot : 1.f;
    float cs  = up + down + fl;
    float cdn = cs > 0.f ? cs : 1.f;
    float pr  = (n_sky > 1.f) ? (float)(mx - mn) / 88.f : 0.f;
    float avg = (n_sky > 0.f) ? sm / fmaxf(n_sky, 1.f) / 127.f : 0.f;
    float* row = melF + b * 128;
    for (int i = 0; i < 25; ++i) row[i] = ih[i] / dn;
    row[25] = up / cdn; row[26] = down / cdn; row[27] = fl / cdn;
    row[28] = pr; row[29] = avg;
    for (int i = 30; i < 128; ++i) row[i] = 0.f;
  }

  // ---- rhythm serial scan (lane 0, wave 1) ----
  if (tid == 32) {
    float rh[32]; for (int i = 0; i < 32; ++i) rh[i] = 0.f;
    float n = 0.f, s = 0.f, s2 = 0.f;
    int prev = 0; bool has = false;
    for (int i = 0; i < S_LEN; ++i) {
      int t = tok[i];
      if (t >= 506 && t < 594) {
        int tm = tms[i];
        if (has) {
          float x = fmaxf((float)(tm - prev), 1.0f);
          int bin = (int)floorf(log10f(x) * 8.0f);
          bin = bin < 0 ? 0 : (bin > 31 ? 31 : bin);
          rh[bin] += 1.f; n += 1.f; s += x; s2 += x * x;
        }
        prev = tm; has = true;
      }
    }
    bool ok = n >= 1.f;
    float dvn  = fmaxf(n, 1.f);
    float mean = s / dvn;
    float var  = fmaxf(s2 / dvn - mean * mean, 0.f);
    float sync = ok ? fminf(sqrtf(var) / (mean + 1e-6f), 10.f) / 10.f : 0.f;
    float hdn  = n > 0.f ? n : 1.f;
    float* row = rhyF + b * 64;
    float ent = 0.f;
    for (int i = 0; i < 32; ++i) {
      float rv = rh[i] / hdn;
      row[i] = rv;
      float p = rv + 1e-10f;
      ent += p * log2f(p);
    }
    ent = ok ? -ent / 5.0f : 0.f;   // log2(32) == 5
    row[32] = sync; row[33] = ent;
    for (int i = 34; i < 64; ++i) row[i] = 0.f;
  }

  // ---- harmony finalize (lane 0, wave 2) ----
  if (tid == 64) {
    float hist[12]; float tot = 0.f;
    bool present[12]; int cs[12]; int run = 0;
    for (int i = 0; i < 12; ++i) {
      hist[i] = pc_hist[i]; tot += hist[i];
      present[i] = hist[i] > 0.f;
      run += present[i] ? 1 : 0; cs[i] = run;
    }
    int c3 = 0, c4 = 0;
    for (int i = 0; i < 12; ++i)
      for (int j = i + 1; j < 12; ++j) {
        int d = j - i;
        if (d == 3 || d == 4) {
          if (present[i] && present[j] && (cs[j - 1] - cs[i]) == 0) {
            if (d == 3) c3++; else c4++;
          }
        }
      }
    int n_unique = run;
    bool cond = (s_npitch >= 3.f) && (n_unique >= 3);
    bool t0 = cond && (c3 >= 1) && (c4 >= 1);
    bool t2 = cond && !t0 && (c3 >= 2);
    bool t3 = cond && !t0 && !t2 && (c4 >= 2);
    bool adj = false;
    for (int i = 0; i < 11; ++i) adj = adj || (present[i] && present[i + 1]);
    float dn = tot > 0.f ? tot : 1.f;
    float* row = harmF + b * 128;
    for (int i = 0; i < 12; ++i) row[i] = hist[i] / dn;
    for (int i = 12; i < 36; ++i) row[i] = 0.f;
    row[12 + 0] = t0 ? 1.f : 0.f;
    row[12 + 2] = t2 ? 1.f : 0.f;
    row[12 + 3] = t3 ? 1.f : 0.f;
    for (int i = 36; i < 48; ++i) row[i] = 0.f;
    row[37] = adj ? 1.f : 0.f;
    for (int i = 48; i < 128; ++i) row[i] = 0.f;
  }

  // ---- dynamics finalize (lane 0, wave 3) ----
  if (tid == 96) {
    float n = s_nvel;
    bool okv = n > 0.f;
    float dv = fmaxf(n, 1.f);
    float vmean = s_vsum / dv;
    float vvar  = fmaxf(s_vsumsq / dv - vmean * vmean, 0.f);
    float dn = n > 0.f ? n : 1.f;
    float* row = dynF + b * 64;
    for (int i = 0; i < 32; ++i) row[i] = vhist[i] / dn;
    row[32] = okv ? (float)(s_vmax - s_vmin) / 127.f : 0.f;
    row[33] = okv ? vmean / 127.f : 0.f;
    row[34] = okv ? sqrtf(vvar) / 127.f : 0.f;
    for (int i = 35; i < 64; ++i) row[i] = 0.f;
  }
}

// ---------------------------------------------------------------------------
// Kernel 2: one modality MLP: h = LN(X@w1+b1)*g+beta; gelu; out = h@w2+b2,
// written into Xcat[:, colOff:colOff+128]. Single block of 512 threads =
// 16 waves, each wave owns one 16x16 tile (2 M-tiles x 8 N-tiles) using
// V_WMMA_F32_16X16X4_F32.
// ---------------------------------------------------------------------------
__global__ __launch_bounds__(512) void mlp_kernel(
    const float* __restrict__ X, int ind,
    const float* __restrict__ w1, const float* __restrict__ b1,
    const float* __restrict__ g,  const float* __restrict__ beta,
    const float* __restrict__ w2, const float* __restrict__ b2,
    float* __restrict__ Xcat, int colOff) {
  __shared__ float Xs[32][128];
  __shared__ float Hs[32][128];
  __shared__ float mu[32];
  __shared__ float rs[32];

  const int tid   = threadIdx.x;
  const int lane  = tid & 31;
  const int wave  = tid >> 5;
  const int mtile = wave >> 3;   // 0..1
  const int ntile = wave & 7;    // 0..7
  const int mrow  = lane & 15;
  const int khalf = (lane < 16) ? 0 : 2;
  const int mbase = mtile * 16 + ((lane < 16) ? 0 : 8);
  const int ncol  = ntile * 16 + mrow;

  for (int idx = tid; idx < 32 * ind; idx += 512)
    Xs[idx / ind][idx % ind] = X[idx];
  __syncthreads();

  // ---- GEMM1: h = X @ w1 + b1 ----
  v8f acc = {0.f, 0.f, 0.f, 0.f, 0.f, 0.f, 0.f, 0.f};
  for (int kb = 0; kb < (ind >> 2); ++kb) {
    int k0 = kb * 4 + khalf;
    v2f a, bb;
    a.x  = Xs[mtile * 16 + mrow][k0];
    a.y  = Xs[mtile * 16 + mrow][k0 + 1];
    bb.x = w1[k0 * 128 + ncol];
    bb.y = w1[(k0 + 1) * 128 + ncol];
    acc = __builtin_amdgcn_wmma_f32_16x16x4_f32(false, a, false, bb,
                                                (short)0, acc, false, false);
  }
  {
    float bias = b1[ncol];
#pragma unroll
    for (int r = 0; r < 8; ++r) Hs[mbase + r][ncol] = acc[r] + bias;
  }
  __syncthreads();

  // ---- layernorm stats ----
  if (tid < 32) {
    float s = 0.f;
    for (int c = 0; c < 128; ++c) s += Hs[tid][c];
    float m = s * (1.0f / 128.0f);
    float v = 0.f;
    for (int c = 0; c < 128; ++c) { float d = Hs[tid][c] - m; v += d * d; }
    v *= (1.0f / 128.0f);
    mu[tid] = m;
    rs[tid] = rsqrtf(v + 1e-5f);
  }
  __syncthreads();

  // ---- normalize + exact GELU ----
  for (int idx = tid; idx < 32 * 128; idx += 512) {
    int r = idx >> 7, c = idx & 127;
    float h = (Hs[r][c] - mu[r]) * rs[r] * g[c] + beta[c];
    h = 0.5f * h * (1.0f + erff(h * 0.70710678118654752f));
    Hs[r][c] = h;
  }
  __syncthreads();

  // ---- GEMM2: out = h @ w2 + b2 ----
  v8f acc2 = {0.f, 0.f, 0.f, 0.f, 0.f, 0.f, 0.f, 0.f};
  for (int kb = 0; kb < 32; ++kb) {
    int k0 = kb * 4 + khalf;
    v2f a, bb;
    a.x  = Hs[mtile * 16 + mrow][k0];
    a.y  = Hs[mtile * 16 + mrow][k0 + 1];
    bb.x = w2[k0 * 128 + ncol];
    bb.y = w2[(k0 + 1) * 128 + ncol];
    acc2 = __builtin_amdgcn_wmma_f32_16x16x4_f32(false, a, false, bb,
                                                 (short)0, acc2, false, false);
  }
  {
    float bias = b2[ncol];
#pragma unroll
    for (int r = 0; r < 8; ++r)
      Xcat[(mbase + r) * 512 + colOff + ncol] = acc2[r] + bias;
  }
}

// ---------------------------------------------------------------------------
// Kernel 3: fusion Out(32x512) = Xcat(32x512) @ fus_w(512x512) + fus_b.
// 8 blocks x 8 waves = 64 waves, one 16x16 tile each (2 M-tiles x 32 N-tiles).
// ---------------------------------------------------------------------------
__global__ __launch_bounds__(256) void fuse_kernel(
    const float* __restrict__ Xcat, const float* __restrict__ fw,
    const float* __restrict__ fb, float* __restrict__ out) {
  const int tid   = threadIdx.x;
  const int lane  = tid & 31;
  const int gwave = blockIdx.x * 8 + (tid >> 5);  // 0..63
  const int mtile = gwave >> 5;                   // 0..1
  const int ntile = gwave & 31;                   // 0..31
  const int mrow  = lane & 15;
  const int khalf = (lane < 16) ? 0 : 2;
  const int mbase = mtile * 16 + ((lane < 16) ? 0 : 8);
  const int ncol  = ntile * 16 + mrow;

  v8f acc = {0.f, 0.f, 0.f, 0.f, 0.f, 0.f, 0.f, 0.f};
  for (int kb = 0; kb < 128; ++kb) {
    int k0 = kb * 4 + khalf;
    v2f a, bb;
    a.x  = Xcat[(mtile * 16 + mrow) * 512 + k0];
    a.y  = Xcat[(mtile * 16 + mrow) * 512 + k0 + 1];
    bb.x = fw[k0 * 512 + ncol];
    bb.y = fw[(k0 + 1) * 512 + ncol];
    acc = __builtin_amdgcn_wmma_f32_16x16x4_f32(false, a, false, bb,
                                                (short)0, acc, false, false);
  }
  float bias = fb[ncol];
#pragma unroll
  for (int r = 0; r < 8; ++r)
    out[(mbase + r) * 512 + ncol] = acc[r] + bias;
}

// ---------------------------------------------------------------------------
extern "C" void kernel_launch(void* const* d_in, const int* in_sizes, int n_in,
                              void* d_out, int out_size, void* d_ws, size_t ws_size,
                              hipStream_t stream) {
  (void)in_sizes; (void)n_in; (void)out_size; (void)ws_size;
  const int* tokens = (const int*)d_in[0];
  const float* f[27];
  for (int i = 1; i < 27; ++i) f[i] = (const float*)d_in[i];

  float* ws    = (float*)d_ws;
  float* harmF = ws;                    // 32*128
  float* melF  = harmF + 32 * 128;      // 32*128
  float* rhyF  = melF  + 32 * 128;      // 32*64
  float* dynF  = rhyF  + 32 * 64;       // 32*64
  float* Xcat  = dynF  + 32 * 64;       // 32*512

  feat_kernel<<<32, 256, 0, stream>>>(tokens, harmF, melF, rhyF, dynF);

  mlp_kernel<<<1, 512, 0, stream>>>(harmF, 128, f[1],  f[2],  f[3],  f[4],  f[5],  f[6],  Xcat, 0);
  mlp_kernel<<<1, 512, 0, stream>>>(melF,  128, f[7],  f[8],  f[9],  f[10], f[11], f[12], Xcat, 128);
  mlp_kernel<<<1, 512, 0, stream>>>(rhyF,   64, f[13], f[14], f[15], f[16], f[17], f[18], Xcat, 256);
  mlp_kernel<<<1, 512, 0, stream>>>(dynF,   64, f[19], f[20], f[21], f[22], f[23], f[24], Xcat, 384);

  fuse_kernel<<<8, 256, 0, stream>>>(Xcat, f[25], f[26], (float*)d_out);
}